// EdgeConv_8761733284511
// MI455X (gfx1250) — compile-verified
//
#include <hip/hip_runtime.h>

typedef float v2f __attribute__((ext_vector_type(2)));
typedef float v8f __attribute__((ext_vector_type(8)));

#define B_ 8
#define C_ 64
#define N_ 4096
#define O_ 64
#define K_ 20
#define NEG_INF (-3.4e38f)

// ---------------------------------------------------------------------------
// 1) squared norms xx[b,n] = sum_c x[b,c,n]^2
// ---------------------------------------------------------------------------
__global__ void k_xx(const float* __restrict__ x, float* __restrict__ xx) {
    int b = blockIdx.y;
    int n = blockIdx.x * 256 + threadIdx.x;
    const float* xb = x + (size_t)b * C_ * N_;
    float s = 0.f;
#pragma unroll
    for (int c = 0; c < C_; ++c) {
        float v = xb[(size_t)c * N_ + n];
        s = fmaf(v, v, s);
    }
    xx[(size_t)b * N_ + n] = s;
}

// ---------------------------------------------------------------------------
// 2) projections via fp32 WMMA:
//    P[b,n,o] = sum_c W[o,c]         * x[b,c,n]
//    Q[b,n,o] = sum_c (W[o,64+c]-W[o,c]) * x[b,c,n]
//    One wave computes a 16-column tile across all 64 output channels.
// ---------------------------------------------------------------------------
__global__ void __launch_bounds__(256) k_proj(const float* __restrict__ x,
                                              const float* __restrict__ W,
                                              float* __restrict__ P,
                                              float* __restrict__ Q) {
    int b = blockIdx.y;
    int wave = threadIdx.x >> 5;
    int lane = threadIdx.x & 31;
    int m16 = lane & 15;
    int hl = lane >> 4;       // which half of the wave
    int khalf = hl * 2;       // K-pair select within 16x16x4 fragment
    int nbase = (blockIdx.x * 8 + wave) * 16;
    int ncol = nbase + m16;
    const float* xb = x + (size_t)b * C_ * N_;

    v8f zero = {0.f, 0.f, 0.f, 0.f, 0.f, 0.f, 0.f, 0.f};
    v8f accP[4], accQ[4];
#pragma unroll
    for (int t = 0; t < 4; ++t) { accP[t] = zero; accQ[t] = zero; }

#pragma unroll
    for (int ks = 0; ks < 16; ++ks) {
        int c = ks * 4 + khalf;
        v2f bf;
        bf.x = xb[(size_t)c * N_ + ncol];
        bf.y = xb[(size_t)(c + 1) * N_ + ncol];
#pragma unroll
        for (int mt = 0; mt < 4; ++mt) {
            int mrow = mt * 16 + m16;
            v2f a1 = *(const v2f*)(W + mrow * 128 + c);
            v2f a2 = *(const v2f*)(W + mrow * 128 + 64 + c);
            a2.x -= a1.x;
            a2.y -= a1.y;
            accP[mt] = __builtin_amdgcn_wmma_f32_16x16x4_f32(
                false, a1, false, bf, (short)0, accP[mt], false, false);
            accQ[mt] = __builtin_amdgcn_wmma_f32_16x16x4_f32(
                false, a2, false, bf, (short)0, accQ[mt], false, false);
        }
    }
    // C layout: lane holds column n = nbase + (lane&15); VGPR r -> o = mt*16 + hl*8 + r
    size_t rowoff = ((size_t)b * N_ + ncol) * O_;
#pragma unroll
    for (int mt = 0; mt < 4; ++mt) {
        int ob = mt * 16 + hl * 8;
        *(v8f*)(P + rowoff + ob) = accP[mt];
        *(v8f*)(Q + rowoff + ob) = accQ[mt];
    }
}

// ---------------------------------------------------------------------------
// 3) Gram matrix via fp32 WMMA with fused per-row top-20 selection.
//    Block = 4 waves, handles 16 query rows of one batch.
//    Orientation: queries on the WMMA N axis (one query per lane),
//    candidates on the M axis (8 per lane in the accumulator VGPRs).
//    Each lane keeps its own sorted top-20 (dist,idx) list in LDS.
// ---------------------------------------------------------------------------
__global__ void __launch_bounds__(128) k_gram_topk(const float* __restrict__ x,
                                                   const float* __restrict__ xx,
                                                   int* __restrict__ idxout) {
    __shared__ float sd[4 * 32 * 21];
    __shared__ int si[4 * 32 * 21];

    int b = blockIdx.y;
    int rowbase = blockIdx.x * 16;
    int wave = threadIdx.x >> 5;
    int lane = threadIdx.x & 31;
    int m16 = lane & 15;
    int hl = lane >> 4;
    int khalf = hl * 2;
    const float* xb = x + (size_t)b * C_ * N_;
    const float* xxb = xx + (size_t)b * N_;

    // preload B fragments (queries), K = 64 in 16 steps of 4
    v2f bf[16];
    int qcol = rowbase + m16;
#pragma unroll
    for (int ks = 0; ks < 16; ++ks) {
        int c = ks * 4 + khalf;
        bf[ks].x = xb[(size_t)c * N_ + qcol];
        bf[ks].y = xb[(size_t)(c + 1) * N_ + qcol];
    }
    float xxq = xxb[qcol];

    // per-lane sorted list (descending), stride 21 avoids LDS bank conflicts
    int lbase = (wave * 32 + lane) * 21;
#pragma unroll
    for (int i = 0; i < 21; ++i) { sd[lbase + i] = NEG_INF; si[lbase + i] = 0; }
    float lmin = NEG_INF;

    int cstart = wave * 1024;  // 4 waves split the 4096 candidate columns
    for (int ct = 0; ct < 64; ++ct) {
        int cb = cstart + ct * 16;
        int ccol = cb + m16;
        v2f af[16];
#pragma unroll
        for (int ks = 0; ks < 16; ++ks) {
            int c = ks * 4 + khalf;
            af[ks].x = xb[(size_t)c * N_ + ccol];
            af[ks].y = xb[(size_t)(c + 1) * N_ + ccol];
        }
        v8f acc = {0.f, 0.f, 0.f, 0.f, 0.f, 0.f, 0.f, 0.f};
#pragma unroll
        for (int ks = 0; ks < 16; ++ks) {
            acc = __builtin_amdgcn_wmma_f32_16x16x4_f32(
                false, af[ks], false, bf[ks], (short)0, acc, false, false);
        }
        // lane holds query qcol; acc[r] = dot(x_q, x_m), m = cb + hl*8 + r
        int moff = cb + hl * 8;
#pragma unroll
        for (int r = 0; r < 8; ++r) {
            int mi = moff + r;
            float pd = fmaf(2.f, acc[r], -xxq) - xxb[mi];  // -(||q-m||^2)
            if (pd > lmin) {
                int p = 19;
                while (p > 0 && sd[lbase + p - 1] < pd) {
                    sd[lbase + p] = sd[lbase + p - 1];
                    si[lbase + p] = si[lbase + p - 1];
                    --p;
                }
                sd[lbase + p] = pd;
                si[lbase + p] = mi;
                lmin = sd[lbase + 19];
            }
        }
    }
    __syncthreads();

    // merge the 8 sorted lists (4 waves x 2 half-wave lists) per query row
    if (threadIdx.x < 16) {
        int q = threadIdx.x;
        int bases[8];
        int h[8];
#pragma unroll
        for (int w = 0; w < 4; ++w) {
            bases[w * 2] = (w * 32 + q) * 21;
            bases[w * 2 + 1] = (w * 32 + 16 + q) * 21;
        }
#pragma unroll
        for (int l = 0; l < 8; ++l) h[l] = 0;
        int* op = idxout + ((size_t)b * N_ + rowbase + q) * K_;
        for (int k = 0; k < K_; ++k) {
            float best = NEG_INF;
            int bl = 0;
#pragma unroll
            for (int l = 0; l < 8; ++l) {
                float d = sd[bases[l] + h[l]];
                if (d > best) { best = d; bl = l; }
            }
            op[k] = si[bases[bl] + h[bl]];
            h[bl]++;
        }
    }
}

// ---------------------------------------------------------------------------
// 4) zero the BN stat accumulators (graph-safe, deterministic each launch)
// ---------------------------------------------------------------------------
__global__ void k_zero(float* s) { s[threadIdx.x] = 0.f; }

// ---------------------------------------------------------------------------
// 5) BN statistics: per-channel sum and sum-of-squares of y = P[j] + Q[n]
// ---------------------------------------------------------------------------
__global__ void __launch_bounds__(256) k_stats(const float* __restrict__ P,
                                               const float* __restrict__ Q,
                                               const int* __restrict__ idx,
                                               float* __restrict__ stats) {
    int b = blockIdx.y;
    int o = threadIdx.x & 63;
    int sub = threadIdx.x >> 6;
    int nb = blockIdx.x * 128;
    float s1 = 0.f, s2 = 0.f;
    for (int i = 0; i < 32; ++i) {
        int n = nb + i * 4 + sub;
        float qv = Q[((size_t)b * N_ + n) * O_ + o];
        const int* ip = idx + ((size_t)b * N_ + n) * K_;
#pragma unroll 4
        for (int k = 0; k < K_; ++k) {
            int j = ip[k];
            float v = P[((size_t)b * N_ + j) * O_ + o] + qv;
            s1 += v;
            s2 = fmaf(v, v, s2);
        }
    }
    __shared__ float r1[256], r2[256];
    r1[threadIdx.x] = s1;
    r2[threadIdx.x] = s2;
    __syncthreads();
    if (sub == 0) {
        s1 = r1[o] + r1[o + 64] + r1[o + 128] + r1[o + 192];
        s2 = r2[o] + r2[o + 64] + r2[o + 128] + r2[o + 192];
        atomicAdd(&stats[o], s1);
        atomicAdd(&stats[64 + o], s2);
    }
}

// ---------------------------------------------------------------------------
// 6) finalize BN: scale = gamma*rsqrt(var+eps), shift = beta - mean*scale
// ---------------------------------------------------------------------------
__global__ void k_finalize(const float* __restrict__ stats,
                           const float* __restrict__ gamma,
                           const float* __restrict__ beta,
                           float* __restrict__ ss) {
    int o = threadIdx.x;
    if (o < 64) {
        const float cnt = (float)B_ * (float)N_ * (float)K_;
        float mean = stats[o] / cnt;
        float var = stats[64 + o] / cnt - mean * mean;
        float inv = rsqrtf(var + 1e-5f);
        float sc = gamma[o] * inv;
        ss[o] = sc;
        ss[64 + o] = beta[o] - mean * sc;
    }
}

// ---------------------------------------------------------------------------
// 7) output: out[b,o,n] = leaky(max_k (scale*(P[j]+Q[n]) + shift))
//    (leaky-ReLU is monotone increasing, so it commutes with max over k)
// ---------------------------------------------------------------------------
__global__ void __launch_bounds__(256) k_out(const float* __restrict__ P,
                                             const float* __restrict__ Q,
                                             const int* __restrict__ idx,
                                             const float* __restrict__ ss,
                                             float* __restrict__ out) {
    int b = blockIdx.y;
    int o = threadIdx.x & 63;
    int sub = threadIdx.x >> 6;
    float scale = ss[o];
    float shift = ss[64 + o];
    for (int i = 0; i < 16; ++i) {
        int n = blockIdx.x * 64 + i * 4 + sub;
        float qv = Q[((size_t)b * N_ + n) * O_ + o];
        const int* ip = idx + ((size_t)b * N_ + n) * K_;
        float m = NEG_INF;
#pragma unroll 4
        for (int k = 0; k < K_; ++k) {
            int j = ip[k];
            float z = fmaf(scale, P[((size_t)b * N_ + j) * O_ + o] + qv, shift);
            m = fmaxf(m, z);
        }
        out[((size_t)b * O_ + o) * N_ + n] = m >= 0.f ? m : 0.2f * m;
    }
}

// ---------------------------------------------------------------------------
// launcher
// ---------------------------------------------------------------------------
extern "C" void kernel_launch(void* const* d_in, const int* in_sizes, int n_in,
                              void* d_out, int out_size, void* d_ws, size_t ws_size,
                              hipStream_t stream) {
    const float* x = (const float*)d_in[0];      // (8, 64, 4096)
    const float* W = (const float*)d_in[1];      // (64, 128)
    const float* gamma = (const float*)d_in[2];  // (64,)
    const float* beta = (const float*)d_in[3];   // (64,)
    float* out = (float*)d_out;                  // (8, 64, 4096)

    // workspace layout
    float* xx = (float*)d_ws;                      // B*N            = 32768 f
    float* P = xx + (size_t)B_ * N_;               // B*N*O          = 2097152 f
    float* Q = P + (size_t)B_ * N_ * O_;           // B*N*O          = 2097152 f
    int* idx = (int*)(Q + (size_t)B_ * N_ * O_);   // B*N*K          = 655360 i
    float* stats = (float*)(idx + (size_t)B_ * N_ * K_);  // 128 f
    float* ss = stats + 128;                       // 128 f

    k_xx<<<dim3(N_ / 256, B_), 256, 0, stream>>>(x, xx);
    k_proj<<<dim3(N_ / 128, B_), 256, 0, stream>>>(x, W, P, Q);
    k_gram_topk<<<dim3(N_ / 16, B_), 128, 0, stream>>>(x, xx, idx);
    k_zero<<<1, 128, 0, stream>>>(stats);
    k_stats<<<dim3(N_ / 128, B_), 256, 0, stream>>>(P, Q, idx, stats);
    k_finalize<<<1, 64, 0, stream>>>(stats, gamma, beta, ss);
    k_out<<<dim3(N_ / 64, B_), 256, 0, stream>>>(P, Q, idx, ss, out);
}